// GraphTransformerLayer_6382321402042
// MI455X (gfx1250) — compile-verified
//
#include <hip/hip_runtime.h>
#include <hip/hip_bf16.h>
#include <math.h>

// ---------------------------------------------------------------------------
// GraphTransformer layer for MI455X (gfx1250, wave32).
// GEMMs: V_WMMA_F32_16X16X4_F32 (fp32-exact matrix path), double-buffered LDS
// pipeline, K-pair-interleaved B staging so fragments are single ds_load_b64.
// Edge softmax/aggregation: scatter-atomic segment ops (HBM-bound, ~3GB traffic).
// ---------------------------------------------------------------------------

typedef float v2f __attribute__((ext_vector_type(2)));
typedef float v8f __attribute__((ext_vector_type(8)));

__device__ __forceinline__ float warp_sum32(float v) {
#pragma unroll
  for (int off = 16; off > 0; off >>= 1) v += __shfl_xor(v, off, 32);
  return v;
}

// Ordered-int trick for float atomic max (init value must be -inf).
__device__ __forceinline__ void atomic_max_float(float* addr, float val) {
  if (val >= 0.0f) atomicMax((int*)addr, __float_as_int(val));
  else             atomicMin((unsigned int*)addr, __float_as_uint(val));
}

// ---------------------------------------------------------------------------
// Generic GEMM: C[M,Ncols] = act(A[M,K] @ B[K,Ncols] + bias)
// Block: 256 threads (8 waves). Block tile 128x64, K-chunk 16, double-buffered.
// Wave w: rows [16w,16w+16), 4 WMMA col-tiles of 16. ACT: 0=none, 1=exact GELU.
// Fragment layouts for V_WMMA_F32_16X16X4_F32 (ISA 7.12.2):
//   A (16x4): lane<16: M=lane, K={0,1}; lane>=16: M=lane-16, K={2,3}
//   B (4x16): symmetric; C/D v8f: row = g + 8*(lane/16), col = lane%16
// B is staged as K-pair-interleaved float2 so each fragment is one ds_load_b64.
// ---------------------------------------------------------------------------
#define ASTRIDE 20   // floats per A row in LDS (16 + pad)
#define BSTRIDE 65   // float2 per B pair-row in LDS (64 + pad)

template <int ACT>
__global__ __launch_bounds__(256) void gemm_wmma_f32(
    const float* __restrict__ A, const float* __restrict__ B,
    const float* __restrict__ bias, float* __restrict__ C,
    int M, int Ncols, int K) {
  __shared__ __align__(16) float As[2][128 * ASTRIDE];
  __shared__ __align__(16) float Bs[2][8 * BSTRIDE * 2];  // 8 K-pairs x 64 cols (float2)

  const int tid  = threadIdx.x;
  const int wave = tid >> 5;
  const int lane = tid & 31;
  const int l16  = lane & 15;
  const int half = lane >> 4;
  const int rowBlock = blockIdx.x * 128;
  const int colBlock = blockIdx.y * 64;

  // Staging coordinates (per thread).
  const int ar = tid >> 1;              // A row within tile (0..127)
  const int ac = (tid & 1) * 8;         // A col base (0 or 8)
  int agr = rowBlock + ar;              // clamped global A row (tail-safe)
  if (agr > M - 1) agr = M - 1;
  const float* aptr = A + (size_t)agr * K + ac;
  const int bp = tid >> 5;              // B K-pair (0..7)
  const int bc = (tid & 31) * 2;        // B col base (0..62 step 2)
  const float* bptr = B + (size_t)(2 * bp) * Ncols + colBlock + bc;

  v8f acc[4];
#pragma unroll
  for (int t = 0; t < 4; ++t)
#pragma unroll
    for (int g = 0; g < 8; ++g) acc[t][g] = 0.0f;

  float4 a0, a1;
  float2 b0, b1;
  auto load_gl = [&](int k0) {
    a0 = *(const float4*)(aptr + k0);
    a1 = *(const float4*)(aptr + k0 + 4);
    const float* bk = bptr + (size_t)k0 * Ncols;
    b0 = *(const float2*)bk;
    b1 = *(const float2*)(bk + Ncols);
  };
  auto store_lds = [&](int buf) {
    float* ad = &As[buf][ar * ASTRIDE + ac];
    *(float4*)ad = a0;
    *(float4*)(ad + 4) = a1;
    float2* bd = (float2*)&Bs[buf][(bp * BSTRIDE + bc) * 2];
    bd[0] = make_float2(b0.x, b1.x);   // {B[2p][c], B[2p+1][c]}
    bd[1] = make_float2(b0.y, b1.y);
  };

  load_gl(0);
  store_lds(0);

  const int nch = K >> 4;
  const int arow = wave * 16 + l16;
  for (int ch = 0; ch < nch; ++ch) {
    const int cur = ch & 1;
    if (ch + 1 < nch) load_gl((ch + 1) * 16);   // prefetch next chunk (in flight)
    __syncthreads();
#pragma unroll
    for (int kk = 0; kk < 16; kk += 4) {
      v2f a;
      const float* ap = &As[cur][arow * ASTRIDE + kk + 2 * half];
      a.x = ap[0];
      a.y = ap[1];
      const int p = (kk >> 1) + half;           // K-pair row for this half
#pragma unroll
      for (int t = 0; t < 4; ++t) {
        const float2 bb = *(const float2*)&Bs[cur][(p * BSTRIDE + t * 16 + l16) * 2];
        v2f b;
        b.x = bb.x;
        b.y = bb.y;
        acc[t] = __builtin_amdgcn_wmma_f32_16x16x4_f32(
            false, a, false, b, (short)0, acc[t], false, false);
      }
    }
    if (ch + 1 < nch) store_lds(cur ^ 1);       // safe: readers passed the barrier
  }

  // Epilogue: bias + optional GELU(exact erf), guarded store.
#pragma unroll
  for (int t = 0; t < 4; ++t) {
    const int col = colBlock + t * 16 + l16;
    const float bv = bias ? bias[col] : 0.0f;
#pragma unroll
    for (int g = 0; g < 8; ++g) {
      const int row = rowBlock + wave * 16 + half * 8 + g;
      if (row < M) {
        float val = acc[t][g] + bv;
        if (ACT == 1) val = 0.5f * val * (1.0f + erff(val * 0.70710678118654752f));
        C[(size_t)row * Ncols + col] = val;
      }
    }
  }
}

// ---------------------------------------------------------------------------
// Edge pass 1: logits[e,h] = dot16(q[dst], k[src]+eproj[e]) / 4 ; atomic-max m.
// One wave per edge: lane owns 4 channels, head = lane/4, 4-lane shuffle reduce.
// ---------------------------------------------------------------------------
__global__ __launch_bounds__(256) void edge_logits_kernel(
    const float* __restrict__ q, const float* __restrict__ k,
    const float* __restrict__ eproj, const long long* __restrict__ eidx,
    float* __restrict__ logits, float* __restrict__ mmax, int E) {
  const int wave = threadIdx.x >> 5, lane = threadIdx.x & 31;
  const long long ed = (long long)blockIdx.x * 8 + wave;
  if (ed >= E) return;
  const int src = (int)eidx[ed];
  const int dst = (int)eidx[(long long)E + ed];
  const int c0 = lane * 4;
  float4 qd = *(const float4*)(q + (size_t)dst * 128 + c0);
  float4 kj = *(const float4*)(k + (size_t)src * 128 + c0);
  float4 ej = *(const float4*)(eproj + (size_t)ed * 128 + c0);
  float p = qd.x * (kj.x + ej.x) + qd.y * (kj.y + ej.y) +
            qd.z * (kj.z + ej.z) + qd.w * (kj.w + ej.w);
  p += __shfl_xor(p, 1, 32);
  p += __shfl_xor(p, 2, 32);
  if ((lane & 3) == 0) {
    const int hh = lane >> 2;
    const float lg = p * 0.25f;  // 1/sqrt(16)
    logits[(size_t)ed * 8 + hh] = lg;
    atomic_max_float(&mmax[(size_t)dst * 8 + hh], lg);
  }
}

// Edge pass 2: ex = exp(logit - m[dst]); denom[dst] += ex (in-place on logits).
__global__ __launch_bounds__(256) void edge_exp_kernel(
    float* __restrict__ logits, const float* __restrict__ mmax,
    float* __restrict__ denom, const long long* __restrict__ eidx, int E) {
  const long long i = (long long)blockIdx.x * 256 + threadIdx.x;
  if (i >= (long long)E * 8) return;
  const long long ed = i >> 3;
  const int hh = (int)(i & 7);
  const int dst = (int)eidx[(long long)E + ed];
  const float ex = __expf(logits[i] - mmax[(size_t)dst * 8 + hh]);
  logits[i] = ex;
  atomicAdd(&denom[(size_t)dst * 8 + hh], ex);
}

// Edge pass 3: agg[dst] += alpha[e,h] * (v[src]+eproj[e]). One wave per edge.
__global__ __launch_bounds__(256) void edge_agg_kernel(
    const float* __restrict__ v, const float* __restrict__ eproj,
    const float* __restrict__ ex, const float* __restrict__ denom,
    const long long* __restrict__ eidx, float* __restrict__ agg, int E) {
  const int wave = threadIdx.x >> 5, lane = threadIdx.x & 31;
  const long long ed = (long long)blockIdx.x * 8 + wave;
  if (ed >= E) return;
  const int src = (int)eidx[ed];
  const int dst = (int)eidx[(long long)E + ed];
  const int hh = lane >> 2;
  const float a = ex[(size_t)ed * 8 + hh] / denom[(size_t)dst * 8 + hh];
  const int c0 = lane * 4;
  float4 vj = *(const float4*)(v + (size_t)src * 128 + c0);
  float4 ej = *(const float4*)(eproj + (size_t)ed * 128 + c0);
  float* out = agg + (size_t)dst * 128 + c0;
  atomicAdd(out + 0, a * (vj.x + ej.x));
  atomicAdd(out + 1, a * (vj.y + ej.y));
  atomicAdd(out + 2, a * (vj.z + ej.z));
  atomicAdd(out + 3, a * (vj.w + ej.w));
}

// ---------------------------------------------------------------------------
// Fused add(2 or 3 inputs) + LayerNorm over D=128. One wave per row (4 f32/lane).
// ---------------------------------------------------------------------------
__global__ __launch_bounds__(256) void add_ln_kernel(
    const float* __restrict__ a, const float* __restrict__ b,
    const float* __restrict__ c, const float* __restrict__ gamma,
    const float* __restrict__ beta, float* __restrict__ out, int M) {
  const int wave = threadIdx.x >> 5, lane = threadIdx.x & 31;
  const long long row = (long long)blockIdx.x * 8 + wave;
  if (row >= M) return;
  const size_t base = (size_t)row * 128 + lane * 4;
  float4 s = *(const float4*)(a + base);
  float4 t = *(const float4*)(b + base);
  s.x += t.x; s.y += t.y; s.z += t.z; s.w += t.w;
  if (c) {
    float4 u = *(const float4*)(c + base);
    s.x += u.x; s.y += u.y; s.z += u.z; s.w += u.w;
  }
  const float mean = warp_sum32(s.x + s.y + s.z + s.w) * 0.0078125f;
  const float d0 = s.x - mean, d1 = s.y - mean, d2 = s.z - mean, d3 = s.w - mean;
  const float var = warp_sum32(d0 * d0 + d1 * d1 + d2 * d2 + d3 * d3) * 0.0078125f;
  const float rstd = rsqrtf(var + 1e-5f);
  const int col = lane * 4;
  out[base + 0] = d0 * rstd * gamma[col + 0] + beta[col + 0];
  out[base + 1] = d1 * rstd * gamma[col + 1] + beta[col + 1];
  out[base + 2] = d2 * rstd * gamma[col + 2] + beta[col + 2];
  out[base + 3] = d3 * rstd * gamma[col + 3] + beta[col + 3];
}

// Init: agg=0, denom=0, m=-inf (required by the ordered-int atomic-max trick).
__global__ __launch_bounds__(256) void init_kernel(
    float* __restrict__ mmax, float* __restrict__ denom,
    float* __restrict__ agg, long long nh, long long nd) {
  const long long i = (long long)blockIdx.x * 256 + threadIdx.x;
  if (i < nd) agg[i] = 0.0f;
  if (i < nh) { mmax[i] = __int_as_float(0xFF800000); denom[i] = 0.0f; }
}

// ---------------------------------------------------------------------------
extern "C" void kernel_launch(void* const* d_in, const int* in_sizes, int n_in,
                              void* d_out, int out_size, void* d_ws, size_t ws_size,
                              hipStream_t stream) {
  (void)n_in; (void)out_size; (void)ws_size;
  const float*     x     = (const float*)d_in[0];
  const long long* eidx  = (const long long*)d_in[1];   // int64 [2,E]
  const float*     eattr = (const float*)d_in[2];
  const float *Wq = (const float*)d_in[3],  *bq = (const float*)d_in[4];
  const float *Wk = (const float*)d_in[5],  *bk = (const float*)d_in[6];
  const float *Wv = (const float*)d_in[7],  *bv = (const float*)d_in[8];
  const float *We = (const float*)d_in[9];
  const float *Wsk = (const float*)d_in[10], *bsk = (const float*)d_in[11];
  const float *g1 = (const float*)d_in[12], *b1 = (const float*)d_in[13];
  const float *g2 = (const float*)d_in[14], *b2 = (const float*)d_in[15];
  const float *Wf1 = (const float*)d_in[16], *bf1 = (const float*)d_in[17];
  const float *Wf2 = (const float*)d_in[18], *bf2 = (const float*)d_in[19];

  const int N  = in_sizes[0] / 128;
  const int E  = in_sizes[1] / 2;
  const int ED = in_sizes[2] / E;   // 64

  // Workspace layout (floats).
  float* ws = (float*)d_ws;
  float* q_    = ws;                         // N*128
  float* k_    = q_    + (size_t)N * 128;    // N*128
  float* v_    = k_    + (size_t)N * 128;    // N*128
  float* skip_ = v_    + (size_t)N * 128;    // N*128
  float* h_    = skip_ + (size_t)N * 128;    // N*128
  float* agg_  = h_    + (size_t)N * 128;    // N*128
  float* f2_   = agg_  + (size_t)N * 128;    // N*128
  float* m_    = f2_   + (size_t)N * 128;    // N*8
  float* den_  = m_    + (size_t)N * 8;      // N*8
  float* t_    = den_  + (size_t)N * 8;      // N*512
  float* lg_   = t_    + (size_t)N * 512;    // E*8   (logits -> ex, in place)
  float* ep_   = lg_   + (size_t)E * 8;      // E*128 (edge projection)

  const dim3 blk(256);

  // 0) init segment-softmax state
  {
    const long long nd = (long long)N * 128, nh = (long long)N * 8;
    init_kernel<<<(unsigned)((nd + 255) / 256), blk, 0, stream>>>(m_, den_, agg_, nh, nd);
  }

  auto gemm = [&](const float* A, const float* B, const float* bias, float* C,
                  int M, int Ncols, int K, bool gelu) {
    dim3 grid((M + 127) / 128, Ncols / 64);
    if (gelu) gemm_wmma_f32<1><<<grid, blk, 0, stream>>>(A, B, bias, C, M, Ncols, K);
    else      gemm_wmma_f32<0><<<grid, blk, 0, stream>>>(A, B, bias, C, M, Ncols, K);
  };

  // 1) projections (fp32 WMMA)
  gemm(x, Wq,  bq,  q_,    N, 128, 128, false);
  gemm(x, Wk,  bk,  k_,    N, 128, 128, false);
  gemm(x, Wv,  bv,  v_,    N, 128, 128, false);
  gemm(x, Wsk, bsk, skip_, N, 128, 128, false);
  gemm(eattr, We, nullptr, ep_, E, 128, ED, false);

  // 2) segment softmax attention over edges
  edge_logits_kernel<<<(E + 7) / 8, blk, 0, stream>>>(q_, k_, ep_, eidx, lg_, m_, E);
  {
    const long long tot = (long long)E * 8;
    edge_exp_kernel<<<(unsigned)((tot + 255) / 256), blk, 0, stream>>>(lg_, m_, den_, eidx, E);
  }
  edge_agg_kernel<<<(E + 7) / 8, blk, 0, stream>>>(v_, ep_, lg_, den_, eidx, agg_, E);

  // 3) h = LN(agg + skip + x)
  add_ln_kernel<<<(N + 7) / 8, blk, 0, stream>>>(agg_, skip_, x, g1, b1, h_, N);

  // 4) FFN + final LN
  gemm(h_, Wf1, bf1, t_,  N, 512, 128, true);    // GELU epilogue
  gemm(t_, Wf2, bf2, f2_, N, 128, 512, false);
  add_ln_kernel<<<(N + 7) / 8, blk, 0, stream>>>(f2_, h_, nullptr, g2, b2,
                                                 (float*)d_out, N);
}